// RegionContraLoss_27822798143931
// MI455X (gfx1250) — compile-verified
//
#include <hip/hip_runtime.h>
#include <hip/hip_bf16.h>
#include <math.h>

typedef __attribute__((ext_vector_type(2))) float v2f;
typedef __attribute__((ext_vector_type(8))) float v8f;

#define GLOBAL_TAU 0.02f
#define LOCAL_TAU  0.5f
#define POS_THRESHOLD 0.3f
#define EPS 1e-7f

#define B_ 64
#define N_ 64
#define S_ 1024
#define D_ 128
#define APAD 132   // padded row stride (floats) for the LDS A tile: bank-conflict free

// ---------------------------------------------------------------------------
// Kernel 1: qn[b,d] = (sum_n q[b,n,d]) / n_valid[b].  Also zeroes the output.
// ---------------------------------------------------------------------------
__global__ __launch_bounds__(256) void prep_kernel(const float* __restrict__ q,
                                                   float* __restrict__ qn,
                                                   float* __restrict__ out) {
    __shared__ float tile[N_ * D_];   // 32 KB
    __shared__ int cnt;
    const int b = blockIdx.x;
    const int t = threadIdx.x;
    if (t == 0) cnt = 0;
    for (int i = t; i < N_ * D_; i += 256)
        tile[i] = q[(long)b * (N_ * D_) + i];
    __syncthreads();
    if (t < N_) {
        float s = 0.f;
        for (int k = 0; k < D_; ++k) s += fabsf(tile[t * D_ + k]);
        if (s > 0.f) atomicAdd(&cnt, 1);
    }
    __syncthreads();
    if (t < D_) {
        float s = 0.f;
        for (int r = 0; r < N_; ++r) s += tile[r * D_ + t];
        qn[b * D_ + t] = s / (float)cnt;   // fold 1/n_valid into qn
    }
    if (b == 0 && t == 0) out[0] = 0.f;
}

// ---------------------------------------------------------------------------
// Kernel 2: WMMA f32 GEMM (64 x 65536 x 128), A staged in LDS (padded),
//   B registers reused across all 4 M-tiles (4 WMMAs per K-step).
//   Per tile: max over 16 s-columns -> partial[b][c][stile];
//   diagonal (c inside M tile) -> local_scores[c][s].
// ---------------------------------------------------------------------------
__global__ __launch_bounds__(128) void gemm_max_kernel(const float* __restrict__ qn,
                                                       const float* __restrict__ doc,
                                                       float* __restrict__ partial,
                                                       float* __restrict__ lscores) {
    __shared__ float sA[B_ * APAD];   // 33 KB: full 64x128 A matrix, padded rows

    const int tid = threadIdx.x;
    // Stage all of A once per block (rows padded to APAD floats).
    for (int i = tid; i < B_ * D_; i += 128)
        sA[(i >> 7) * APAD + (i & 127)] = qn[i];
    __syncthreads();

    const int lane  = tid & 31;
    const int wave  = tid >> 5;
    const int ntile = blockIdx.x * 4 + wave;   // 0..4095 over flattened (c, s/16)
    const int nl    = lane & 15;               // A: M-lane, B: N-lane
    const int khalf = lane >> 4;               // 0: K=k0+{0,1}, 1: K=k0+{2,3}
    const int c     = ntile >> 6;              // doc batch index
    const int stile = ntile & 63;              // s/16 within c

    const float* brow = doc + ((long)ntile * 16 + nl) * D_ + 2 * khalf;
    const float* arow = sA + nl * APAD + 2 * khalf;

    v8f acc0 = {}, acc1 = {}, acc2 = {}, acc3 = {};
    for (int k0 = 0; k0 < D_; k0 += 4) {
        v2f b = *(const v2f*)(brow + k0);
        v2f a0 = *(const v2f*)(arow + 0  * 16 * APAD + k0);
        v2f a1 = *(const v2f*)(arow + 1  * 16 * APAD + k0);
        v2f a2 = *(const v2f*)(arow + 2  * 16 * APAD + k0);
        v2f a3 = *(const v2f*)(arow + 3  * 16 * APAD + k0);
        acc0 = __builtin_amdgcn_wmma_f32_16x16x4_f32(false, a0, false, b, (short)0, acc0, false, false);
        acc1 = __builtin_amdgcn_wmma_f32_16x16x4_f32(false, a1, false, b, (short)0, acc1, false, false);
        acc2 = __builtin_amdgcn_wmma_f32_16x16x4_f32(false, a2, false, b, (short)0, acc2, false, false);
        acc3 = __builtin_amdgcn_wmma_f32_16x16x4_f32(false, a3, false, b, (short)0, acc3, false, false);
    }

    const int mdiag = c >> 4;      // which M tile holds the diagonal for this c
    const int mrow  = c & 15;      // row within that tile

    #pragma unroll
    for (int mt = 0; mt < 4; ++mt) {
        v8f acc = (mt == 0) ? acc0 : (mt == 1) ? acc1 : (mt == 2) ? acc2 : acc3;
        float av[8];
        #pragma unroll
        for (int j = 0; j < 8; ++j) av[j] = acc[j];

        // max over the 16 s-columns: reduce within each 16-lane half
        #pragma unroll
        for (int j = 0; j < 8; ++j) {
            float v = av[j];
            #pragma unroll
            for (int off = 8; off >= 1; off >>= 1)
                v = fmaxf(v, __shfl_xor(v, off, 16));
            if (nl == 0) {
                int brow_idx = mt * 16 + j + 8 * khalf;   // b
                partial[((long)brow_idx * B_ + c) * 64 + stile] = v;
            }
        }

        // diagonal tile: write local_scores[c, s]
        if (mt == mdiag && khalf == (mrow >> 3))
            lscores[(long)c * S_ + stile * 16 + nl] = av[mrow & 7];
    }
}

// ---------------------------------------------------------------------------
// Block reduction helpers
// ---------------------------------------------------------------------------
__device__ inline float blkSumN(float v, float* sm, int half) {
    int t = threadIdx.x;
    sm[t] = v; __syncthreads();
    for (int s = half; s > 0; s >>= 1) { if (t < s) sm[t] += sm[t + s]; __syncthreads(); }
    float r = sm[0]; __syncthreads(); return r;
}
__device__ inline float blkMaxN(float v, float* sm, int half) {
    int t = threadIdx.x;
    sm[t] = v; __syncthreads();
    for (int s = half; s > 0; s >>= 1) { if (t < s) sm[t] = fmaxf(sm[t], sm[t + s]); __syncthreads(); }
    float r = sm[0]; __syncthreads(); return r;
}

// ---------------------------------------------------------------------------
// Kernel 3: global InfoNCE term. Block per b, thread per c.
// ---------------------------------------------------------------------------
__global__ __launch_bounds__(64) void global_loss_kernel(const float* __restrict__ partial,
                                                         float* __restrict__ out) {
    __shared__ float sm[64];
    __shared__ float sdiag;
    const int b = blockIdx.x, c = threadIdx.x;
    const float* p = partial + ((long)b * B_ + c) * 64;
    float m = -INFINITY;
    for (int t = 0; t < 64; ++t) m = fmaxf(m, p[t]);
    float l = m * (1.0f / GLOBAL_TAU);
    if (c == b) sdiag = l;
    float rowmax = blkMaxN(l, sm, 32);
    float se = blkSumN(expf(l - rowmax), sm, 32);
    if (c == 0)
        atomicAdd(out, -(sdiag - rowmax - logf(se)) * (1.0f / (float)B_));
}

// ---------------------------------------------------------------------------
// Kernel 4: local InfoNCE with bbox / self masks. One block per b.
// ---------------------------------------------------------------------------
__global__ __launch_bounds__(256) void local_loss_kernel(const float* __restrict__ ls_all,
                                                         const unsigned char* __restrict__ bbox,
                                                         float* __restrict__ out) {
    __shared__ float sm[256];
    const int b = blockIdx.x, t = threadIdx.x;
    float myls[4], mybb[4];
    #pragma unroll
    for (int i = 0; i < 4; ++i) {
        int s = t * 4 + i;
        myls[i] = ls_all[(long)b * S_ + s];
        mybb[i] = (float)bbox[(long)b * S_ + s];
    }
    float cnt = mybb[0] + mybb[1] + mybb[2] + mybb[3];
    cnt = blkSumN(cnt, sm, 128);
    const bool valid = cnt > 0.f;

    float lg[4], lmax = -INFINITY;
    #pragma unroll
    for (int i = 0; i < 4; ++i) { lg[i] = myls[i] * (1.0f / LOCAL_TAU) + EPS; lmax = fmaxf(lmax, lg[i]); }
    lmax = blkMaxN(lmax, sm, 128);

    float se = 0.f;
    #pragma unroll
    for (int i = 0; i < 4; ++i) se += expf(lg[i] - lmax);
    se = blkSumN(se, sm, 128);
    const float logZ = logf(se + EPS);

    float s1 = 0.f, s2 = 0.f;
    #pragma unroll
    for (int i = 0; i < 4; ++i) {
        float pm = valid ? mybb[i] : (myls[i] > POS_THRESHOLD ? 1.f : 0.f);
        s1 += pm * (lg[i] - lmax - logZ);
        s2 += pm;
    }
    s1 = blkSumN(s1, sm, 128);
    s2 = blkSumN(s2, sm, 128);
    if (t == 0) atomicAdd(out, -(s1 / (s2 + EPS)) * (1.0f / (float)B_));
}

// ---------------------------------------------------------------------------
extern "C" void kernel_launch(void* const* d_in, const int* in_sizes, int n_in,
                              void* d_out, int out_size, void* d_ws, size_t ws_size,
                              hipStream_t stream) {
    const float* q   = (const float*)d_in[0];          // (64,64,128) f32
    const float* doc = (const float*)d_in[1];          // (64,1024,128) f32
    const unsigned char* bbox = (const unsigned char*)d_in[2]; // (64,1024) bool
    float* out = (float*)d_out;                        // scalar f32

    char* ws = (char*)d_ws;
    float* qn      = (float*)ws;                               // 32 KB
    float* partial = (float*)(ws + 32768);                     // 1 MB (64*64*64)
    float* lsc     = (float*)(ws + 32768 + 1048576);           // 256 KB (64*1024)

    prep_kernel<<<B_, 256, 0, stream>>>(q, qn, out);
    gemm_max_kernel<<<1024, 128, 0, stream>>>(qn, doc, partial, lsc);
    global_loss_kernel<<<B_, 64, 0, stream>>>(partial, out);
    local_loss_kernel<<<B_, 256, 0, stream>>>(lsc, bbox, out);
}